// IouOverlapLoss_6708738917083
// MI455X (gfx1250) — compile-verified
//
#include <hip/hip_runtime.h>
#include <stdint.h>

// ---------------- problem constants ----------------
#define BB     16
#define HH     512
#define WW     512
#define HW     (HH*WW)          // 262144
#define NP     12544
#define NS     37632            // NP * 3 oversampled
#define NUNC   9408
#define NRAND  3136
#define TOTPTS (NS + NRAND)     // 40768
#define CH2    40               // point-loss chunks of 1024 per batch
#define CHB    64               // bbox blocks per batch (1024 thr x 4 px)
#define CH4    64               // region blocks per batch (4096 px each)

typedef unsigned int u32;
typedef float v2f __attribute__((ext_vector_type(2)));
typedef float v8f __attribute__((ext_vector_type(8)));

// ---------------- CDNA5 helpers ----------------

// Full wave32 sum using the matrix pipe:
// A(16x4): vgpr0 = v  -> A[m,0]=v[m], A[m,2]=v[m+16]; vgpr1 = 0.
// B(4x16) = ones => D[m,n] = v[m] + v[m+16].
// D vgpr j holds D[j,lane] (lanes0-15) / D[8+j,lane-16] (lanes16-31),
// so sum of the 8 D vgprs = half-wave sum; one xor-16 shuffle completes it.
__device__ __forceinline__ float wave_sum(float v) {
  v2f a; a[0] = v;   a[1] = 0.0f;
  v2f b; b[0] = 1.0f; b[1] = 1.0f;
  v8f c = {};
  c = __builtin_amdgcn_wmma_f32_16x16x4_f32(false, a, false, b,
                                            (short)0, c, false, false);
  float s = c[0]+c[1]+c[2]+c[3]+c[4]+c[5]+c[6]+c[7];
  s += __shfl_xor(s, 16, 32);
  return s;
}

// Async copy 16B/lane global -> LDS (gfx1250 ASYNCcnt path).
__device__ __forceinline__ void async_ld_f4(float* lds_dst, const float* gsrc) {
  unsigned l = (unsigned)(uintptr_t)lds_dst;              // low 32 bits = LDS offset
  unsigned long long g = (unsigned long long)(uintptr_t)gsrc;
  asm volatile("global_load_async_to_lds_b128 %0, %1, off"
               :: "v"(l), "v"(g) : "memory");
}
__device__ __forceinline__ void async_wait0() {
  asm volatile("s_wait_asynccnt 0" ::: "memory");
}

// ---------------- math helpers ----------------
__device__ __forceinline__ float fetch(const float* img, int yy, int xx) {
  if ((unsigned)yy < (unsigned)HH && (unsigned)xx < (unsigned)WW)
    return img[(yy << 9) + xx];
  return 0.0f;
}

// grid_sample bilinear, align_corners=False, zero padding (matches reference)
__device__ __forceinline__ float bilinear(const float* img, float cx, float cy) {
  float x = cx * (float)WW - 0.5f;
  float y = cy * (float)HH - 0.5f;
  float xf = floorf(x), yf = floorf(y);
  int x0 = (int)xf, y0 = (int)yf;
  float wx1 = x - xf, wx0 = 1.0f - wx1;
  float wy1 = y - yf, wy0 = 1.0f - wy1;
  float v00 = fetch(img, y0,     x0    );
  float v01 = fetch(img, y0,     x0 + 1);
  float v10 = fetch(img, y0 + 1, x0    );
  float v11 = fetch(img, y0 + 1, x0 + 1);
  return v00*(wy0*wx0) + v01*(wy0*wx1) + v10*(wy1*wx0) + v11*(wy1*wx1);
}

__device__ __forceinline__ float log_sigmoid(float x) {
  return (x >= 0.0f) ? -log1pf(expf(-x)) : (x - log1pf(expf(x)));
}

// ---------------- kernels ----------------

__global__ void k_init(u32* hist, u32* state, int* bbox) {
  int i = blockIdx.x * blockDim.x + threadIdx.x;
  if (i < BB * 256) hist[i] = 0u;
  if (i < BB) {
    state[i*4+0] = 0u;            // selected prefix (tau being built)
    state[i*4+1] = (u32)NUNC;     // k remaining
    state[i*4+2] = 0u;            // m (taken among equals)
    state[i*4+3] = 1u;            // count of equals
    bbox[i*8+0] = WW;             // xmin
    bbox[i*8+1] = -1;             // xmax  (<0 => empty)
    bbox[i*8+2] = HH;             // ymin
    bbox[i*8+3] = -1;             // ymax
  }
}

// Sample pred at oversampled coords; key = bits of |pl| (monotone for >=0 floats)
__global__ void k_sample(const float* __restrict__ pred,
                         const float* __restrict__ cos_,
                         u32* __restrict__ keys, float* __restrict__ plbuf) {
  int i = blockIdx.x * 256 + threadIdx.x;
  if (i >= BB * NS) return;
  __builtin_prefetch(&cos_[2*i + 4096], 0, 0);   // global_prefetch_b8
  int b = i / NS;
  float cx = cos_[2*i], cy = cos_[2*i + 1];
  float pl = bilinear(pred + (size_t)b * HW, cx, cy);
  keys[i]  = __float_as_uint(fabsf(pl));
  plbuf[i] = pl;
}

// Radix-select pass: histogram of 8-bit digit among keys matching the prefix
__global__ void k_hist(const u32* __restrict__ keys, const u32* __restrict__ state,
                       u32* __restrict__ hist, int shift) {
  int i = blockIdx.x * 256 + threadIdx.x;
  if (i >= BB * NS) return;
  int b = i / NS;
  u32 key = keys[i];
  u32 hi_mask = (shift >= 24) ? 0u : (~0u << (shift + 8));
  if ((key & hi_mask) == (state[b*4] & hi_mask))
    atomicAdd(&hist[b*256 + ((key >> shift) & 0xFF)], 1u);
}

// Pick the digit bucket containing the k-th smallest; update prefix / k_rem
__global__ void k_pick(u32* __restrict__ hist, u32* __restrict__ state, int shift) {
  int b = blockIdx.x;
  __shared__ u32 h[256];
  h[threadIdx.x] = hist[b*256 + threadIdx.x];
  __syncthreads();
  if (threadIdx.x == 0) {
    u32 krem = state[b*4+1];
    u32 cum = 0, c = 0; int chosen = 255;
    for (int bin = 0; bin < 256; ++bin) {
      c = h[bin];
      if (cum + c >= krem) { chosen = bin; break; }
      cum += c;
    }
    state[b*4+0] |= ((u32)chosen) << shift;
    state[b*4+1]  = krem - cum;
    if (shift == 0) {
      state[b*4+2] = krem - cum;            // m equals to take
      state[b*4+3] = (c > 0u) ? c : 1u;     // total equals
    }
  }
  __syncthreads();
  hist[b*256 + threadIdx.x] = 0u;           // ready for next pass
}

// Point-sampled CE + dice partial sums (selected oversampled pts + rand pts)
__global__ void k_point(const float* __restrict__ pred, const float* __restrict__ gt,
                        const float* __restrict__ cos_, const float* __restrict__ crand,
                        const u32* __restrict__ keys, const float* __restrict__ plbuf,
                        const u32* __restrict__ state, float* __restrict__ part1) {
  int b  = blockIdx.x / CH2;
  int ch = blockIdx.x % CH2;
  int q  = ch * 1024 + threadIdx.x;

  float ce = 0.f, sp = 0.f, st = 0.f, spt = 0.f;
  if (q < TOTPTS) {
    float w = 0.f, z = 0.f, t = 0.f;
    if (q < NS) {
      int gi = b * NS + q;
      u32 key = keys[gi];
      u32 tau = state[b*4+0];
      if (key <= tau) {
        w = (key < tau) ? 1.0f
                        : ((float)state[b*4+2] / (float)state[b*4+3]);
        z = plbuf[gi];
        t = bilinear(gt + (size_t)b * HW, cos_[2*gi], cos_[2*gi+1]);
      }
    } else {
      int gi = b * NRAND + (q - NS);
      float cx = crand[2*gi], cy = crand[2*gi+1];
      w = 1.0f;
      z = bilinear(pred + (size_t)b * HW, cx, cy);
      t = bilinear(gt   + (size_t)b * HW, cx, cy);
    }
    if (w > 0.0f) {
      float az = fabsf(z);
      float cv = fmaxf(z, 0.0f) - z * t + log1pf(expf(-az));
      float pv = 1.0f / (1.0f + expf(-z));
      ce = w * cv; sp = w * pv; st = w * t; spt = w * pv * t;
    }
  }

  float r0 = wave_sum(ce), r1 = wave_sum(sp), r2 = wave_sum(st), r3 = wave_sum(spt);
  __shared__ float red[32][4];
  int wv = threadIdx.x >> 5, ln = threadIdx.x & 31;
  if (ln == 0) { red[wv][0]=r0; red[wv][1]=r1; red[wv][2]=r2; red[wv][3]=r3; }
  __syncthreads();
  if (threadIdx.x == 0) {
    float a0=0.f,a1=0.f,a2=0.f,a3=0.f;
    for (int w2 = 0; w2 < 32; ++w2) {
      a0 += red[w2][0]; a1 += red[w2][1]; a2 += red[w2][2]; a3 += red[w2][3];
    }
    float* o = part1 + (size_t)(b*CH2 + ch) * 4;
    o[0]=a0; o[1]=a1; o[2]=a2; o[3]=a3;
  }
}

// Bounding box of sigmoid(pred)>0.5 via int min/max atomics (deterministic)
__global__ void k_bbox(const float* __restrict__ pred, int* __restrict__ bbox) {
  int b   = blockIdx.x / CHB;
  int blk = blockIdx.x % CHB;
  int base = blk * 4096;
  int xmn = WW, xmx = -1, ymn = HH, ymx = -1;
  for (int k = 0; k < 4; ++k) {
    int p = base + k * 1024 + threadIdx.x;          // pixel in image
    float z = pred[(size_t)b * HW + p];
    float s = 1.0f / (1.0f + expf(-z));
    if (s > 0.5f) {
      int x = p & (WW - 1), y = p >> 9;
      xmn = min(xmn, x); xmx = max(xmx, x);
      ymn = min(ymn, y); ymx = max(ymx, y);
    }
  }
  __shared__ int sxmn[1024], sxmx[1024], symn[1024], symx[1024];
  sxmn[threadIdx.x]=xmn; sxmx[threadIdx.x]=xmx;
  symn[threadIdx.x]=ymn; symx[threadIdx.x]=ymx;
  __syncthreads();
  for (int s2 = 512; s2 > 0; s2 >>= 1) {
    if (threadIdx.x < (unsigned)s2) {
      sxmn[threadIdx.x] = min(sxmn[threadIdx.x], sxmn[threadIdx.x+s2]);
      sxmx[threadIdx.x] = max(sxmx[threadIdx.x], sxmx[threadIdx.x+s2]);
      symn[threadIdx.x] = min(symn[threadIdx.x], symn[threadIdx.x+s2]);
      symx[threadIdx.x] = max(symx[threadIdx.x], symx[threadIdx.x+s2]);
    }
    __syncthreads();
  }
  if (threadIdx.x == 0) {
    atomicMin(&bbox[b*8+0], sxmn[0]);
    atomicMax(&bbox[b*8+1], sxmx[0]);
    atomicMin(&bbox[b*8+2], symn[0]);
    atomicMax(&bbox[b*8+3], symx[0]);
  }
}

// Region-aware BCE + dice partials; tiles staged via async global->LDS copies
__global__ void k_region(const float* __restrict__ pred, const float* __restrict__ gt,
                         const int* __restrict__ bbox, float* __restrict__ part2) {
  int b   = blockIdx.x / CH4;
  int blk = blockIdx.x % CH4;
  size_t gbase = (size_t)b * HW + (size_t)blk * 4096;

  int xmx = bbox[b*8+1], ymx = bbox[b*8+3];
  int xmn = bbox[b*8+0], ymn = bbox[b*8+2];
  bool empty = (xmx < 0);
  int x1 = empty ? 0 : xmn, y1 = empty ? 0 : ymn;
  int x2 = empty ? (WW-1) : xmx, y2 = empty ? (HH-1) : ymx;
  bool valid = (x2 > x1) && (y2 > y1);

  __shared__ float predS[1024];
  __shared__ float gtS[1024];

  float a_w = 0.f, a_bc = 0.f, a_pg = 0.f, a_p = 0.f, a_g = 0.f;
  for (int it = 0; it < 4; ++it) {
    int off = it * 1024 + (int)threadIdx.x * 4;
    async_ld_f4(&predS[threadIdx.x * 4], pred + gbase + off);
    async_ld_f4(&gtS  [threadIdx.x * 4], gt   + gbase + off);
    async_wait0();
    __syncthreads();
    for (int k = 0; k < 4; ++k) {
      int idx = (int)threadIdx.x * 4 + k;
      int p_lin = blk * 4096 + it * 1024 + idx;     // pixel in image
      int x = p_lin & (WW - 1), y = p_lin >> 9;
      float z = predS[idx], g = gtS[idx];
      bool inb = valid && x >= x1 && x <= x2 && y >= y1 && y <= y2;
      float w = inb ? 2.0f : 0.5f;
      float lp  = log_sigmoid(z);
      float l1p = log_sigmoid(-z);
      float bce = -(g * lp + (1.0f - g) * l1p);
      float pv  = 1.0f / (1.0f + expf(-z));
      a_w += w; a_bc += bce * w; a_pg += pv * g * w; a_p += pv * w; a_g += g * w;
    }
    __syncthreads();   // protect LDS before next async overwrite
  }

  float r0 = wave_sum(a_w), r1 = wave_sum(a_bc), r2 = wave_sum(a_pg),
        r3 = wave_sum(a_p), r4 = wave_sum(a_g);
  __shared__ float red[8][5];
  int wv = threadIdx.x >> 5, ln = threadIdx.x & 31;
  if (ln == 0) { red[wv][0]=r0; red[wv][1]=r1; red[wv][2]=r2; red[wv][3]=r3; red[wv][4]=r4; }
  __syncthreads();
  if (threadIdx.x == 0) {
    float s0=0.f,s1=0.f,s2=0.f,s3=0.f,s4=0.f;
    for (int w2 = 0; w2 < 8; ++w2) {
      s0+=red[w2][0]; s1+=red[w2][1]; s2+=red[w2][2]; s3+=red[w2][3]; s4+=red[w2][4];
    }
    float* o = part2 + (size_t)(b*CH4 + blk) * 8;
    o[0]=s0; o[1]=s1; o[2]=s2; o[3]=s3; o[4]=s4;
  }
}

__global__ void k_final(const float* __restrict__ part1, const float* __restrict__ part2,
                        float* __restrict__ out) {
  float lm = 0.f, ld = 0.f, lrb = 0.f, lrd = 0.f;
  for (int b = 0; b < BB; ++b) {
    float ce=0.f, sp=0.f, st=0.f, spt=0.f;
    for (int ch = 0; ch < CH2; ++ch) {
      const float* o = part1 + (size_t)(b*CH2 + ch) * 4;
      ce += o[0]; sp += o[1]; st += o[2]; spt += o[3];
    }
    lm += ce / (float)NP;
    ld += 1.0f - (2.0f * spt + 1.0f) / (sp + st + 1.0f);

    float ws=0.f, bc=0.f, pg=0.f, pw=0.f, gw=0.f;
    for (int blk = 0; blk < CH4; ++blk) {
      const float* o = part2 + (size_t)(b*CH4 + blk) * 8;
      ws += o[0]; bc += o[1]; pg += o[2]; pw += o[3]; gw += o[4];
    }
    lrb += bc / fmaxf(ws, 1e-6f);
    lrd += 1.0f - (2.0f * pg + 1.0f) / (pw + gw + 1.0f);
  }
  out[0] = lm / 16.0f;
  out[1] = ld / 16.0f;
  out[2] = lrb / 16.0f;
  out[3] = lrd / 16.0f;
}

// ---------------- host launcher ----------------
extern "C" void kernel_launch(void* const* d_in, const int* in_sizes, int n_in,
                              void* d_out, int out_size, void* d_ws, size_t ws_size,
                              hipStream_t stream) {
  (void)in_sizes; (void)n_in; (void)out_size; (void)ws_size;
  const float* pred  = (const float*)d_in[0];
  const float* gt    = (const float*)d_in[1];
  const float* cos_  = (const float*)d_in[2];
  const float* crand = (const float*)d_in[3];
  float* out = (float*)d_out;

  char* ws = (char*)d_ws;                       // ~4.88 MB used
  u32*   keys  = (u32*)(ws + 0);                // BB*NS u32  = 2,408,448 B
  float* plbuf = (float*)(ws + 2408448);        // BB*NS f32  = 2,408,448 B
  u32*   hist  = (u32*)(ws + 4816896);          // BB*256 u32 = 16,384 B
  u32*   state = (u32*)(ws + 4833280);          // BB*4 u32
  int*   bbox  = (int*)(ws + 4833792);          // BB*8 i32
  float* part1 = (float*)(ws + 4834304);        // BB*CH2*4 f32 = 10,240 B
  float* part2 = (float*)(ws + 4844544);        // BB*CH4*8 f32 = 32,768 B

  k_init<<<16, 256, 0, stream>>>(hist, state, bbox);
  k_sample<<<(BB*NS + 255)/256, 256, 0, stream>>>(pred, cos_, keys, plbuf);
  for (int shift = 24; shift >= 0; shift -= 8) {
    k_hist<<<(BB*NS + 255)/256, 256, 0, stream>>>(keys, state, hist, shift);
    k_pick<<<BB, 256, 0, stream>>>(hist, state, shift);
  }
  k_point<<<BB*CH2, 1024, 0, stream>>>(pred, gt, cos_, crand, keys, plbuf, state, part1);
  k_bbox<<<BB*CHB, 1024, 0, stream>>>(pred, bbox);
  k_region<<<BB*CH4, 256, 0, stream>>>(pred, gt, bbox, part2);
  k_final<<<1, 1, 0, stream>>>(part1, part2, out);
}